// EnhancedDualGCN_78443282695014
// MI455X (gfx1250) — compile-verified
//
#include <hip/hip_runtime.h>

typedef __attribute__((ext_vector_type(2))) float v2f;
typedef __attribute__((ext_vector_type(8))) float v8f;

#define BN_EPS 1e-5f

__device__ __forceinline__ v8f wmma4(v2f a, v2f b, v8f c) {
  // V_WMMA_F32_16X16X4_F32 : D = A(16x4) x B(4x16) + C(16x16), all fp32
  return __builtin_amdgcn_wmma_f32_16x16x4_f32(
      /*neg_a=*/false, a, /*neg_b=*/false, b,
      /*c_mod=*/(short)0, c, /*reuse_a=*/false, /*reuse_b=*/false);
}

// ---------------------------------------------------------------------------
// z[n, 0:32] = emb[n, :], z[n, 32:64] = nan_to_num(x[n,:]) @ ft_W.T + ft_b
// ---------------------------------------------------------------------------
__global__ void build_z_kernel(const float* __restrict__ x,
                               const float* __restrict__ emb,
                               const float* __restrict__ ftW,
                               const float* __restrict__ ftb,
                               float* __restrict__ z, int n_nodes) {
  int t = blockIdx.x * blockDim.x + threadIdx.x;
  if (t >= n_nodes * 64) return;
  int n = t >> 6;
  int c = t & 63;
  float v;
  if (c < 32) {
    v = emb[(size_t)n * 32 + c];
  } else {
    int cc = c - 32;
    float s = ftb[cc];
    #pragma unroll
    for (int i = 0; i < 16; ++i) {
      float xv = x[(size_t)n * 16 + i];
      xv = __builtin_isnan(xv) ? 0.f : xv;
      s += xv * ftW[cc * 16 + i];
    }
    v = s;
  }
  z[(size_t)n * 64 + c] = v;
}

// ---------------------------------------------------------------------------
// WMMA fp32 GEMM: C[nrows,128] (=|+=) op(A[nrows,K] @ W[0:128, 0:K]^T (+bias))
// W row stride (in floats) is `wstride` so sub-blocks of a wider weight matrix
// can be used (fusion layer = two K=128 GEMMs with wstride 256).
//
// - W is staged once per block into LDS, pre-swizzled into the exact v2f
//   fragment layout WMMA consumes:  lds2[(kb*2+g)*128 + j] = (W[j][kb*4+2g],
//   W[j][kb*4+2g+1]).  64KB for K=128, 32KB for K=64.
// - Each wave owns 16 rows; A panel (K/4 v2f) lives in registers for all 8
//   j-tiles.  Per j-tile: burst-load B fragments from LDS, then run the WMMA
//   chain with two independent accumulators (no load between matrix ops).
// ---------------------------------------------------------------------------
template <int K, bool RELU, bool BIAS, bool ACCUM>
__global__ __launch_bounds__(256, 1)
void wmma_gemm_kernel(const float* __restrict__ A,
                      const float* __restrict__ W, int wstride,
                      const float* __restrict__ bias,
                      float* __restrict__ C, int nrows) {
  constexpr int KB = K / 4;          // wmma steps
  constexpr int KBG = KB * 2;        // v2f rows in LDS
  __shared__ v2f lds2[KBG * 128];

  // cooperative stage of W into fragment layout
  for (int idx = threadIdx.x; idx < KBG * 128; idx += blockDim.x) {
    int kbg = idx >> 7;
    int j = idx & 127;
    int k = (kbg >> 1) * 4 + (kbg & 1) * 2;
    v2f v;
    v.x = W[(size_t)j * wstride + k];
    v.y = W[(size_t)j * wstride + k + 1];
    lds2[idx] = v;
  }
  __syncthreads();

  const int wave = threadIdx.x >> 5;
  const int lane = threadIdx.x & 31;
  const int row0 = (blockIdx.x * (blockDim.x >> 5) + wave) * 16;
  if (row0 >= nrows) return;  // wave-uniform; no further barriers
  const int g = lane >> 4;
  const int l = lane & 15;

  v2f afrag[KB];
  const float* arow = A + (size_t)(row0 + l) * K + 2 * g;
  #pragma unroll
  for (int kb = 0; kb < KB; ++kb) {
    afrag[kb].x = arow[kb * 4 + 0];
    afrag[kb].y = arow[kb * 4 + 1];
  }

  #pragma unroll 1
  for (int jt = 0; jt < 8; ++jt) {
    const int j = jt * 16 + l;
    v2f bfrag[KB];
    #pragma unroll
    for (int kb = 0; kb < KB; ++kb) bfrag[kb] = lds2[(kb * 2 + g) * 128 + j];

    v8f acc0 = {}, acc1 = {};
    #pragma unroll
    for (int kb = 0; kb < KB; kb += 2) {
      acc0 = wmma4(afrag[kb + 0], bfrag[kb + 0], acc0);
      acc1 = wmma4(afrag[kb + 1], bfrag[kb + 1], acc1);
    }

    float bj = BIAS ? bias[j] : 0.f;
    #pragma unroll
    for (int r = 0; r < 8; ++r) {
      float v = acc0[r] + acc1[r] + bj;
      if (RELU) v = fmaxf(v, 0.f);
      size_t off = (size_t)(row0 + r + 8 * g) * 128 + j;
      if (ACCUM) C[off] += v; else C[off] = v;
    }
  }
}

// ---------------------------------------------------------------------------
// Degree / norm helpers
// ---------------------------------------------------------------------------
__global__ void deg_init_kernel(float* __restrict__ deg, int n_nodes) {
  int t = blockIdx.x * blockDim.x + threadIdx.x;
  if (t < n_nodes) deg[t] = 1.0f;  // self-loop weight
}

__global__ void edge_deg_kernel(const int* __restrict__ ei, const float* __restrict__ ew,
                                float* __restrict__ deg, int n_edges) {
  int t = blockIdx.x * blockDim.x + threadIdx.x;
  if (t >= n_edges) return;
  int dst = ei[n_edges + t];
  atomicAdd(&deg[dst], ew[t]);
}

__global__ void dis_kernel(const float* __restrict__ deg, float* __restrict__ dis, int n_nodes) {
  int t = blockIdx.x * blockDim.x + threadIdx.x;
  if (t >= n_nodes) return;
  float d = deg[t];
  dis[t] = d > 0.f ? rsqrtf(d) : 0.f;
}

// agg[n,f] = xw[n,f] * dis[n]^2   (self-loop term)
__global__ void agg_init_kernel(const float* __restrict__ xw, const float* __restrict__ dis,
                                float* __restrict__ agg, int n_nodes) {
  int t = blockIdx.x * blockDim.x + threadIdx.x;
  if (t >= n_nodes * 128) return;
  int n = t >> 7;
  float dn = dis[n];
  agg[t] = xw[t] * dn * dn;
}

// One wave per edge; each lane scatters 4 consecutive features (float4 gather).
__global__ void agg_edges_kernel(const int* __restrict__ ei, const float* __restrict__ ew,
                                 const float* __restrict__ dis, const float* __restrict__ xw,
                                 float* __restrict__ agg, int n_edges) {
  long long t = (long long)blockIdx.x * blockDim.x + threadIdx.x;
  int e = (int)(t >> 5);
  if (e >= n_edges) return;
  int f = ((int)t & 31) * 4;
  int s = ei[e];
  int d = ei[n_edges + e];
  float w = dis[s] * ew[e] * dis[d];
  const float4 xv = *(const float4*)(xw + (size_t)s * 128 + f);
  float* ap = agg + (size_t)d * 128 + f;
  atomicAdd(ap + 0, xv.x * w);
  atomicAdd(ap + 1, xv.y * w);
  atomicAdd(ap + 2, xv.z * w);
  atomicAdd(ap + 3, xv.w * w);
}

// hc += BN(relu(agg + bias))
__global__ void post_kernel(const float* __restrict__ agg, const float* __restrict__ bias,
                            const float* __restrict__ bn, float* __restrict__ hc, int total) {
  int t = blockIdx.x * blockDim.x + threadIdx.x;
  if (t >= total) return;
  int f = t & 127;
  float a = fmaxf(agg[t] + bias[f], 0.f);
  float gmul = bn[f], bb = bn[128 + f], m = bn[256 + f], var = bn[384 + f];
  hc[t] += (a - m) * rsqrtf(var + BN_EPS) * gmul + bb;
}

// hf = BN(relu(hf)) in place
__global__ void bnrelu_kernel(float* __restrict__ hf, const float* __restrict__ bn, int total) {
  int t = blockIdx.x * blockDim.x + threadIdx.x;
  if (t >= total) return;
  int f = t & 127;
  float a = fmaxf(hf[t], 0.f);
  float gmul = bn[f], bb = bn[128 + f], m = bn[256 + f], var = bn[384 + f];
  hf[t] = (a - m) * rsqrtf(var + BN_EPS) * gmul + bb;
}

// out[n] = clip(dot(hf[n,:], linW) + linb, -10, 10); one wave per node
__global__ void final_kernel(const float* __restrict__ hf, const float* __restrict__ linW,
                             const float* __restrict__ linb, float* __restrict__ out, int nrows) {
  int wave = threadIdx.x >> 5;
  int lane = threadIdx.x & 31;
  int n = blockIdx.x * (blockDim.x >> 5) + wave;
  if (n >= nrows) return;
  const float4 h4 = *(const float4*)(hf + (size_t)n * 128 + lane * 4);
  const float4 w4 = *(const float4*)(linW + lane * 4);
  float s = h4.x * w4.x + h4.y * w4.y + h4.z * w4.z + h4.w * w4.w;
  #pragma unroll
  for (int off = 16; off; off >>= 1) s += __shfl_xor(s, off, 32);
  if (lane == 0) out[n] = fminf(fmaxf(s + linb[0], -10.f), 10.f);
}

// ---------------------------------------------------------------------------
extern "C" void kernel_launch(void* const* d_in, const int* in_sizes, int n_in,
                              void* d_out, int out_size, void* d_ws, size_t ws_size,
                              hipStream_t stream) {
  (void)n_in; (void)out_size; (void)ws_size;
  const float* x       = (const float*)d_in[0];
  const int*   ei_call = (const int*)  d_in[1];
  const float* ew_call = (const float*)d_in[2];
  const int*   ei_loc  = (const int*)  d_in[3];
  const float* ew_loc  = (const float*)d_in[4];
  const float* emb     = (const float*)d_in[5];
  const float* ft_W    = (const float*)d_in[6];
  const float* ft_b    = (const float*)d_in[7];
  const float* comb_W  = (const float*)d_in[8];
  const float* comb_b  = (const float*)d_in[9];
  const float* call_W  = (const float*)d_in[10];
  const float* call_b  = (const float*)d_in[11];
  const float* loc_W   = (const float*)d_in[12];
  const float* loc_b   = (const float*)d_in[13];
  const float* fus_W   = (const float*)d_in[14];
  const float* fus_b   = (const float*)d_in[15];
  const float* lin_W   = (const float*)d_in[16];
  const float* lin_b   = (const float*)d_in[17];
  const float* bn_call = (const float*)d_in[18];
  const float* bn_loc  = (const float*)d_in[19];
  const float* bn_fus  = (const float*)d_in[20];

  const int N = in_sizes[0] / 16;
  const int E = in_sizes[2];
  float* out = (float*)d_out;

  // workspace carve-up
  float* ws = (float*)d_ws;
  float* z        = ws; ws += (size_t)N * 64;
  float* B1       = ws; ws += (size_t)N * 128;  // h0 -> xw -> hf
  float* hc_call  = ws; ws += (size_t)N * 128;
  float* hc_loc   = ws; ws += (size_t)N * 128;
  float* agg      = ws; ws += (size_t)N * 128;
  float* deg      = ws; ws += N;
  float* dis_call = ws; ws += N;
  float* dis_loc  = ws; ws += N;

  const int TPB = 256;
  const int grid_rows  = ((N + 15) / 16 + 7) / 8;  // 16 rows/wave, 8 waves/block
  const int grid_nodeF = (int)(((size_t)N * 128 + TPB - 1) / TPB);
  const int grid_node  = (N + TPB - 1) / TPB;
  const int grid_edge  = (E + TPB - 1) / TPB;
  const int grid_scat  = (int)(((long long)E * 32 + TPB - 1) / TPB);
  const int grid_z     = (int)(((size_t)N * 64 + TPB - 1) / TPB);
  const int grid_final = (N + 7) / 8;

  // 1. input transform + concat
  build_z_kernel<<<grid_z, TPB, 0, stream>>>(x, emb, ft_W, ft_b, z, N);

  // 2. h0 = relu(z @ comb_W^T + comb_b)   (WMMA fp32, K=64)
  wmma_gemm_kernel<64, true, true, false><<<grid_rows, TPB, 0, stream>>>(
      z, comb_W, 64, comb_b, B1, N);
  hipMemcpyAsync(hc_call, B1, (size_t)N * 128 * sizeof(float), hipMemcpyDeviceToDevice, stream);
  hipMemcpyAsync(hc_loc,  B1, (size_t)N * 128 * sizeof(float), hipMemcpyDeviceToDevice, stream);

  // 3. symmetric normalization coefficients per branch
  deg_init_kernel<<<grid_node, TPB, 0, stream>>>(deg, N);
  edge_deg_kernel<<<grid_edge, TPB, 0, stream>>>(ei_call, ew_call, deg, E);
  dis_kernel<<<grid_node, TPB, 0, stream>>>(deg, dis_call, N);
  deg_init_kernel<<<grid_node, TPB, 0, stream>>>(deg, N);
  edge_deg_kernel<<<grid_edge, TPB, 0, stream>>>(ei_loc, ew_loc, deg, E);
  dis_kernel<<<grid_node, TPB, 0, stream>>>(deg, dis_loc, N);

  // 4. GCN branches (3 layers each): GEMM -> self-loop init -> edge scatter -> BN/ReLU/residual
  for (int br = 0; br < 2; ++br) {
    const int*   ei  = br == 0 ? ei_call : ei_loc;
    const float* ewp = br == 0 ? ew_call : ew_loc;
    const float* dis = br == 0 ? dis_call : dis_loc;
    const float* Ws  = br == 0 ? call_W : loc_W;
    const float* bs  = br == 0 ? call_b : loc_b;
    const float* bn  = br == 0 ? bn_call : bn_loc;
    float* hc = br == 0 ? hc_call : hc_loc;
    for (int i = 0; i < 3; ++i) {
      wmma_gemm_kernel<128, false, false, false><<<grid_rows, TPB, 0, stream>>>(
          hc, Ws + (size_t)i * 128 * 128, 128, nullptr, B1, N);
      agg_init_kernel<<<grid_nodeF, TPB, 0, stream>>>(B1, dis, agg, N);
      agg_edges_kernel<<<grid_scat, TPB, 0, stream>>>(ei, ewp, dis, B1, agg, E);
      post_kernel<<<grid_nodeF, TPB, 0, stream>>>(agg, bs + (size_t)i * 128, bn, hc, N * 128);
    }
  }

  // 5. fusion: hf = h_call @ W[:, :128]^T ; hf += h_loc @ W[:, 128:]^T + fus_b
  wmma_gemm_kernel<128, false, false, false><<<grid_rows, TPB, 0, stream>>>(
      hc_call, fus_W, 256, nullptr, B1, N);
  wmma_gemm_kernel<128, false, true, true><<<grid_rows, TPB, 0, stream>>>(
      hc_loc, fus_W + 128, 256, fus_b, B1, N);
  bnrelu_kernel<<<grid_nodeF, TPB, 0, stream>>>(B1, bn_fus, N * 128);

  // 6. linear head + clip
  final_kernel<<<grid_final, TPB, 0, stream>>>(B1, lin_W, lin_b, out, N);
}